// PriorityQueue_77103252897911
// MI455X (gfx1250) — compile-verified
//
#include <hip/hip_runtime.h>
#include <math.h>

#define B 64
#define N 1024
#define F 128
#define M 256
#define E 128
#define H 8
#define O 128

typedef float v2f __attribute__((ext_vector_type(2)));
typedef float v8f __attribute__((ext_vector_type(8)));

__device__ __forceinline__ float lrelu(float x) { return x >= 0.f ? x : 0.01f * x; }

// ---- workspace layout (float offsets) -------------------------------------
#define WS_ZSUM   0                        // B*F
#define WS_WV     (WS_ZSUM + B*F)          // B*E
#define WS_ATT1   (WS_WV + B*E)            // B*H*N   layout [B][H][N]
#define WS_ATT2   (WS_ATT1 + B*H*N)        // B*H*M   layout [B][H][M]
#define WS_MEMO   (WS_ATT2 + B*H*M)        // B*M*O
#define WS_IDX    (WS_MEMO + B*M*O)        // B*N ints
// total = 2,834,432 floats ~= 11.3 MB

// ---- kernel 1: zsum[b,f] = sum_n z[b,n,f] ---------------------------------
__global__ void k_zsum(const float* __restrict__ z, float* __restrict__ zsum) {
  int b = blockIdx.x, f = threadIdx.x;                 // 128 threads
  const float* zp = z + (size_t)b * N * F + f;
  float s = 0.f;
  for (int n = 0; n < N; ++n) s += zp[(size_t)n * F];
  zsum[b * F + f] = s;
}

// ---- kernel 2: write_values[b,e] = tanh(zsum[b]·Wv[:,e] + N*bv[e]) --------
__global__ void k_writevals(const float* __restrict__ zsum, const float* __restrict__ Wv,
                            const float* __restrict__ bv, float* __restrict__ wv) {
  int b = blockIdx.x, e = threadIdx.x;                 // 128 threads
  float s = (float)N * bv[e];
  for (int f = 0; f < F; ++f) s += zsum[b * F + f] * Wv[f * E + e];
  wv[b * E + e] = tanhf(s);
}

// ---- kernel 3: att1[b,h,n] = z[b,n]·Wa1[:,h] + ba1[h]  (one wave per (b,n))
__global__ void k_att1(const float* __restrict__ z, const float* __restrict__ Wa1,
                       const float* __restrict__ ba1, float* __restrict__ att1) {
  int tid = threadIdx.x;
  int wave = (blockIdx.x * blockDim.x + tid) >> 5;
  int lane = tid & 31;
  int b = wave >> 10, n = wave & (N - 1);
  const float4 zv = *(const float4*)(z + ((size_t)(b * N + n)) * F + lane * 4);
  const float* wp = Wa1 + (lane * 4) * H;
  float p[H];
  #pragma unroll
  for (int h = 0; h < H; ++h)
    p[h] = zv.x * wp[h] + zv.y * wp[H + h] + zv.z * wp[2 * H + h] + zv.w * wp[3 * H + h];
  #pragma unroll
  for (int h = 0; h < H; ++h)
    for (int off = 16; off; off >>= 1) p[h] += __shfl_xor(p[h], off, 32);
  if (lane == 0) {
    #pragma unroll
    for (int h = 0; h < H; ++h)
      att1[((size_t)b * H + h) * N + n] = p[h] + ba1[h];
  }
}

// ---- kernel 4: att2[b,h,m] = mem[b,m]·Wa2[:,h] + ba2[h] (one wave per (b,m))
__global__ void k_att2(const float* __restrict__ mem, const float* __restrict__ Wa2,
                       const float* __restrict__ ba2, float* __restrict__ att2) {
  int tid = threadIdx.x;
  int wave = (blockIdx.x * blockDim.x + tid) >> 5;
  int lane = tid & 31;
  int b = wave >> 8, m = wave & (M - 1);
  const float4 mv = *(const float4*)(mem + ((size_t)(b * M + m)) * E + lane * 4);
  const float* wp = Wa2 + (lane * 4) * H;
  float p[H];
  #pragma unroll
  for (int h = 0; h < H; ++h)
    p[h] = mv.x * wp[h] + mv.y * wp[H + h] + mv.z * wp[2 * H + h] + mv.w * wp[3 * H + h];
  #pragma unroll
  for (int h = 0; h < H; ++h)
    for (int off = 16; off; off >>= 1) p[h] += __shfl_xor(p[h], off, 32);
  if (lane == 0) {
    #pragma unroll
    for (int h = 0; h < H; ++h)
      att2[((size_t)b * H + h) * M + m] = p[h] + ba2[h];
  }
}

// ---- kernel 5 (WMMA f32): memo[b] = mem[b](256x128) @ Wo(128x128) + bo ----
// One wave computes one 16x16 tile with 32 chained V_WMMA_F32_16X16X4_F32.
// f32 A/B fragment layout (ISA 7.12.2): lane L, VGPR v -> element
//   A[row = L%16][k = v + 2*(L/16)],  B[k = v + 2*(L/16)][col = L%16]
__global__ void k_memo(const float* __restrict__ mem, const float* __restrict__ Wo,
                       const float* __restrict__ bo, float* __restrict__ memo) {
  int tid = threadIdx.x;
  int wave = (blockIdx.x * blockDim.x + tid) >> 5;
  int lane = tid & 31;
  int b = wave >> 7;               // 128 tiles per batch: 16 (M/16) x 8 (O/16)
  int t = wave & 127;
  int m_base = (t >> 3) * 16;
  int n_base = (t & 7) * 16;
  int half = lane >> 4;
  int r = lane & 15;

  v8f acc;
  float bov = bo[n_base + r];      // same output column for all 8 C rows
  #pragma unroll
  for (int i = 0; i < 8; ++i) acc[i] = bov;

  const float* arow = mem + ((size_t)(b * M) + m_base + r) * E + 2 * half;
  const float* bcol = Wo + (size_t)(2 * half) * O + n_base + r;
  #pragma unroll
  for (int kk = 0; kk < E; kk += 4) {
    v2f a;  a.x = arow[kk];               a.y = arow[kk + 1];
    v2f bm; bm.x = bcol[(size_t)kk * O];  bm.y = bcol[(size_t)(kk + 1) * O];
    acc = __builtin_amdgcn_wmma_f32_16x16x4_f32(
        /*neg_a=*/false, a, /*neg_b=*/false, bm,
        /*c_mod=*/(short)0, acc, /*reuse_a=*/false, /*reuse_b=*/false);
  }

  // D layout: VGPR v -> row m_base + v + 8*half, col n_base + r
  float* op = memo + ((size_t)(b * M) + m_base + 8 * half) * O + n_base + r;
  #pragma unroll
  for (int v = 0; v < 8; ++v) op[(size_t)v * O] = acc[v];
}

// ---- kernel 6: fused dual softmax / argmax (one thread per (b,n)) ---------
__global__ void k_attn(const float* __restrict__ att1, const float* __restrict__ att2,
                       const float* __restrict__ bias_mask, const float* __restrict__ Wh,
                       const float* __restrict__ bh, int* __restrict__ idx_out) {
  __shared__ float att2_s[H][M];   // 8 KB
  __shared__ float bias_s[M];      // 1 KB
  int tid = threadIdx.x;                     // 256 threads
  int b = blockIdx.x >> 2;
  int n = ((blockIdx.x & 3) << 8) + tid;

  for (int i = tid; i < H * M; i += 256)
    att2_s[i >> 8][i & (M - 1)] = att2[(size_t)b * H * M + i];
  bias_s[tid] = (bias_mask[b * M + tid] - 1.f) * 1e9f;   // M == blockDim
  __syncthreads();

  float a1[H];
  #pragma unroll
  for (int h = 0; h < H; ++h) a1[h] = att1[((size_t)b * H + h) * N + n];

  float mx[H], wsum[H];
  #pragma unroll
  for (int h = 0; h < H; ++h) {
    float mxh = -3.0e38f;
    for (int m = 0; m < M; ++m) {
      float s = lrelu(a1[h] + att2_s[h][m]) + bias_s[m];
      mxh = fmaxf(mxh, s);
    }
    float sm = 0.f;
    for (int m = 0; m < M; ++m)
      sm += __expf(lrelu(a1[h] + att2_s[h][m]) + bias_s[m] - mxh);
    mx[h] = mxh;
    wsum[h] = Wh[h] / sm;          // fold Wh into the normalizer
  }

  // c[b,n,m] = bh + sum_h Wh[h]*softmax_m(...); argmax over monotone transform
  float bhv = bh[0];
  float best = -3.4e38f; int bidx = 0;
  for (int m = 0; m < M; ++m) {
    float c = bhv;
    float bm = bias_s[m];
    #pragma unroll
    for (int h = 0; h < H; ++h)
      c += wsum[h] * __expf(lrelu(a1[h] + att2_s[h][m]) + bm - mx[h]);
    float val = lrelu(c) + bm;
    if (val > best) { best = val; bidx = m; }   // first max, like jnp.argmax
  }
  idx_out[(size_t)b * N + n] = bidx;
}

// ---- kernel 7: output[b,n,:] = memo[b, idx[b,n], :] -----------------------
__global__ void k_gather(const float* __restrict__ memo, const int* __restrict__ idx,
                         float* __restrict__ out) {
  int bn = blockIdx.x;                       // B*N blocks, 128 threads
  int b = bn >> 10;
  int o = threadIdx.x;
  out[(size_t)bn * O + o] = memo[((size_t)(b * M) + idx[bn]) * O + o];
}

// ---- kernel 8: new_mem / rolled write_mask / clamped bias_mask ------------
__global__ void k_tail(const float* __restrict__ mem, const float* __restrict__ wmask,
                       const float* __restrict__ bmask, const float* __restrict__ wv,
                       float* __restrict__ new_mem, float* __restrict__ new_wmask,
                       float* __restrict__ new_bmask) {
  int b = blockIdx.x, t = threadIdx.x;       // 128 threads (e index)
  float wvv = wv[b * E + t];
  for (int m = 0; m < M; ++m) {
    float wm = wmask[b * M + m];
    size_t o = ((size_t)(b * M) + m) * E + t;
    new_mem[o] = mem[o] + wvv * wm;
  }
  for (int m = t; m < M; m += 128) {
    new_wmask[b * M + m] = wmask[b * M + ((m + M - 1) & (M - 1))];  // roll +1
    new_bmask[b * M + m] = fminf(bmask[b * M + m] + wmask[b * M + m], 1.f);
  }
}

extern "C" void kernel_launch(void* const* d_in, const int* in_sizes, int n_in,
                              void* d_out, int out_size, void* d_ws, size_t ws_size,
                              hipStream_t stream) {
  const float* z     = (const float*)d_in[0];
  const float* mem   = (const float*)d_in[1];
  const float* wmask = (const float*)d_in[2];
  const float* bmask = (const float*)d_in[3];
  const float* Wv    = (const float*)d_in[4];
  const float* bv    = (const float*)d_in[5];
  const float* Wo    = (const float*)d_in[6];
  const float* bo    = (const float*)d_in[7];
  const float* Wa1   = (const float*)d_in[8];
  const float* ba1   = (const float*)d_in[9];
  const float* Wa2   = (const float*)d_in[10];
  const float* ba2   = (const float*)d_in[11];
  const float* Wh    = (const float*)d_in[12];
  const float* bh    = (const float*)d_in[13];

  float* ws   = (float*)d_ws;
  float* zsum = ws + WS_ZSUM;
  float* wv   = ws + WS_WV;
  float* att1 = ws + WS_ATT1;
  float* att2 = ws + WS_ATT2;
  float* memo = ws + WS_MEMO;
  int*   idx  = (int*)(ws + WS_IDX);

  float* out  = (float*)d_out;                    // (B,N,O)
  float* nmem = out + (size_t)B * N * O;          // (B,M,E)
  float* nwm  = nmem + (size_t)B * M * E;         // (B,M)
  float* nbm  = nwm + (size_t)B * M;              // (B,M)

  k_zsum     <<<B,               F,   0, stream>>>(z, zsum);
  k_writevals<<<B,               E,   0, stream>>>(zsum, Wv, bv, wv);
  k_att1     <<<(B * N) / 8,     256, 0, stream>>>(z, Wa1, ba1, att1);
  k_att2     <<<(B * M) / 8,     256, 0, stream>>>(mem, Wa2, ba2, att2);
  k_memo     <<<(B * 128) / 8,   256, 0, stream>>>(mem, Wo, bo, memo);
  k_attn     <<<B * (N / 256),   256, 0, stream>>>(att1, att2, bmask, Wh, bh, idx);
  k_gather   <<<B * N,           O,   0, stream>>>(memo, idx, out);
  k_tail     <<<B,               E,   0, stream>>>(mem, wmask, bmask, wv, nmem, nwm, nbm);
}